// KalmanFilter_16123307229331
// MI455X (gfx1250) — compile-verified
//
#include <hip/hip_runtime.h>
#include <stdint.h>

// Kalman-filter forward scan, T=512 steps, N=8192 independent 5-state filters.
// Key simplification (exact w.r.t. the reference): A has no identity term, so
// Sigma_pred[:,3:5] = [e3,e4] always => K rows 0..2 are 0, K[3:5,:] = inv(I+LL^T),
// and the entire Sigma recursion is dead code for y_hat = mu[0:3].
// => memory-bound streaming recurrence; use CDNA5 async global->LDS pipeline.

#define T_STEPS 512
#define NFILT   8192
#define BLOCK   32           // one wave32 per workgroup: no barriers needed
#define DEPTH   24           // time-steps in flight (2 async ops each -> 48 <= 63)
#define SLOT_F  192          // floats per LDS slot: 640B zl tile + 128B times
#define ZL_BYTES 640
#define TAIL    (T_STEPS - DEPTH)

static_assert(2 * (DEPTH - 1) == 46, "wait immediate must match DEPTH");

// ---------------- CDNA5 async-copy plumbing (probe both toolchains) --------
typedef int v4i __attribute__((ext_vector_type(4)));
typedef __attribute__((address_space(1))) v4i gbl_v4i;
typedef __attribute__((address_space(3))) v4i lds_v4i;

__device__ __forceinline__ lds_v4i* to_lds(void* p) {
  // generic LDS pointer -> addrspace(3): low 32 bits are the LDS offset
  return (lds_v4i*)(uint32_t)(uint64_t)p;
}
__device__ __forceinline__ gbl_v4i* to_gbl(const void* p) {
  return (gbl_v4i*)(uint64_t)p;
}

#if __has_builtin(__builtin_amdgcn_global_load_async_to_lds_b128)
__device__ __forceinline__ void async_copy_b128(const void* g, void* l) {
  __builtin_amdgcn_global_load_async_to_lds_b128(to_gbl(g), to_lds(l), 0, 0);
}
#else
__device__ __forceinline__ void async_copy_b128(const void* g, void* l) {
  uint32_t loff = (uint32_t)(uint64_t)l;
  uint64_t ga   = (uint64_t)g;
  asm volatile("global_load_async_to_lds_b128 %0, %1, off"
               :: "v"(loff), "v"(ga) : "memory");
}
#endif

#if __has_builtin(__builtin_amdgcn_s_wait_asynccnt)
#define WAIT_ASYNC(n) __builtin_amdgcn_s_wait_asynccnt(n)
#else
#define WAIT_ASYNC(n) asm volatile("s_wait_asynccnt %0" :: "i"(n) : "memory")
#endif

#if __has_builtin(__builtin_amdgcn_s_wait_dscnt)
#define WAIT_DS(n) __builtin_amdgcn_s_wait_dscnt(n)
#else
#define WAIT_DS(n) asm volatile("s_wait_dscnt %0" :: "i"(n) : "memory")
#endif

// ---------------------------------------------------------------------------

__global__ __launch_bounds__(BLOCK) void kf_forward_kernel(
    const float* __restrict__ zl,     // (T, N, 5)
    const float* __restrict__ mu0,    // (N, 5)
    const float* __restrict__ times,  // (T, N)
    float* __restrict__ out)          // (T, N, 3)
{
  __shared__ __align__(16) float smem[DEPTH * SLOT_F];

  const int tid = threadIdx.x;           // 0..31
  const int n0  = blockIdx.x * BLOCK;    // first filter of this block
  const int n   = n0 + tid;

  // Issue one step's tile: 640B of zl + 128B of times = 48 x 16B chunks,
  // covered by exactly 2 wave-wide async b128 instructions (2nd half-EXEC).
  auto issue = [&](int p) {
    char* lz = (char*)&smem[(p % DEPTH) * SLOT_F];
    const char* gz = (const char*)zl    + ((size_t)p * NFILT + (size_t)n0) * 20u;
    const char* gt = (const char*)times + ((size_t)p * NFILT + (size_t)n0) * 4u;
    async_copy_b128(gz + tid * 16, lz + tid * 16);           // chunks 0..31
    if (tid < 16) {                                          // chunks 32..47
      const char* g2 = (tid < 8) ? (gz + 512 + tid * 16) : (gt + (tid - 8) * 16);
      char*       l2 = (tid < 8) ? (lz + 512 + tid * 16)
                                 : (lz + ZL_BYTES + (tid - 8) * 16);
      async_copy_b128(g2, l2);
    }
  };

  // State (the only live carry): mu = [x, y, th, v, om] and prev_t.
  float x  = mu0[(size_t)n * 5 + 0];
  float y  = mu0[(size_t)n * 5 + 1];
  float th = mu0[(size_t)n * 5 + 2];
  float v  = mu0[(size_t)n * 5 + 3];
  float om = mu0[(size_t)n * 5 + 4];
  float pt = times[n];                       // times[0, n]  (first dt == 0)

  auto step = [&](int t, float z0, float z1, float h0, float h1, float h2,
                  float tt) {
    float dt  = tt - pt;
    float c   = __cosf(th);
    float s   = __sinf(th);
    // R = L L^T with L = [[e^h0, 0], [h1, e^h2]];  S = I + R;  Si = S^-1
    float l00 = __expf(h0);
    float l11 = __expf(h2);
    float r00 = l00 * l00;
    float r10 = h1 * l00;
    float r11 = fmaf(h1, h1, l11 * l11);
    float S00 = 1.0f + r00;
    float S11 = 1.0f + r11;
    float det = fmaf(S00, S11, -(r10 * r10));
    float id  = 1.0f / det;
    float Si00 = S11 * id, Si01 = -r10 * id, Si11 = S00 * id;
    float in0 = z0 - v;
    float in1 = z1 - om;
    float vn  = v  + fmaf(Si00, in0, Si01 * in1);
    float on  = om + fmaf(Si01, in0, Si11 * in1);
    float thn = fmaf(om, dt, th);
    float vdt = v * dt;
    x = fmaf(vdt, c, x);
    y = fmaf(vdt, s, y);
    float* o = out + ((size_t)t * NFILT + (size_t)n) * 3u;
    __builtin_nontemporal_store(x,   o + 0);
    __builtin_nontemporal_store(y,   o + 1);
    __builtin_nontemporal_store(thn, o + 2);
    th = thn; v = vn; om = on; pt = tt;
  };

  // Prologue: fill the async pipeline DEPTH steps deep.
  for (int p = 0; p < DEPTH; ++p) issue(p);

  int t = 0;
  for (; t < TAIL; ++t) {
    WAIT_ASYNC(46);                          // 2*(DEPTH-1): step-t tile landed
    const float* sp = &smem[(t % DEPTH) * SLOT_F];
    float z0 = sp[tid * 5 + 0];
    float z1 = sp[tid * 5 + 1];
    float h0 = sp[tid * 5 + 2];
    float h1 = sp[tid * 5 + 3];
    float h2 = sp[tid * 5 + 4];
    float tt = sp[160 + tid];
    WAIT_DS(0);                              // LDS reads retired before refill
    issue(t + DEPTH);
    step(t, z0, z1, h0, h1, h2, tt);
  }

  WAIT_ASYNC(0);                             // drain: all remaining tiles in
  for (; t < T_STEPS; ++t) {
    const float* sp = &smem[(t % DEPTH) * SLOT_F];
    float z0 = sp[tid * 5 + 0];
    float z1 = sp[tid * 5 + 1];
    float h0 = sp[tid * 5 + 2];
    float h1 = sp[tid * 5 + 3];
    float h2 = sp[tid * 5 + 4];
    float tt = sp[160 + tid];
    step(t, z0, z1, h0, h1, h2, tt);
  }
}

extern "C" void kernel_launch(void* const* d_in, const int* in_sizes, int n_in,
                              void* d_out, int out_size, void* d_ws,
                              size_t ws_size, hipStream_t stream) {
  (void)in_sizes; (void)n_in; (void)out_size; (void)d_ws; (void)ws_size;
  const float* zl    = (const float*)d_in[0];   // (512, 8192, 5) f32
  const float* mu0   = (const float*)d_in[1];   // (8192, 5) f32
  const float* times = (const float*)d_in[2];   // (512, 8192) f32
  float* out = (float*)d_out;                   // (512, 8192, 3) f32

  dim3 grid(NFILT / BLOCK);   // 256 single-wave workgroups
  dim3 block(BLOCK);
  kf_forward_kernel<<<grid, block, 0, stream>>>(zl, mu0, times, out);
}